// Recommender_62268435857506
// MI455X (gfx1250) — compile-verified
//
#include <hip/hip_runtime.h>
#include <hip/hip_bf16.h>
#include <stdint.h>

#ifndef __has_builtin
#define __has_builtin(x) 0
#endif

// ---------------------------------------------------------------------------
// Problem constants (from the reference)
// ---------------------------------------------------------------------------
#define NU 100000      // users
#define NE 100000      // entities
#define NN 200000      // nodes
#define DD 64          // feature dim
#define EKG 1500000
#define EPREF 1500000
#define NNZI 1000000

typedef unsigned int u32x4 __attribute__((ext_vector_type(4)));
typedef int          i32x4v __attribute__((ext_vector_type(4)));
typedef int          i32x8v __attribute__((ext_vector_type(8)));
typedef float        v2f __attribute__((ext_vector_type(2)));
typedef float        v8f __attribute__((ext_vector_type(8)));

// Native fp32 global atomic add (no-return -> global_atomic_add_f32).
__device__ __forceinline__ void atomic_add_f32(float* p, float v) {
  unsafeAtomicAdd(p, v);
}

// ---------------------------------------------------------------------------
// init: node_res = concat(user0, ent0); user_res = user0   (float4 streams)
// ---------------------------------------------------------------------------
__global__ __launch_bounds__(256) void init_out_kernel(
    const float* __restrict__ user0, const float* __restrict__ ent0,
    float* __restrict__ out_user, float* __restrict__ out_node) {
  int i = blockIdx.x * 256 + threadIdx.x;              // float4 index
  if (i >= NN * (DD / 4)) return;
  float4 v = (i < NU * (DD / 4)) ? ((const float4*)user0)[i]
                                 : ((const float4*)ent0)[i - NU * (DD / 4)];
  ((float4*)out_node)[i] = v;
  if (i < NU * (DD / 4)) ((float4*)out_user)[i] = v;
}

// ---------------------------------------------------------------------------
// Edge scatter: one wave per edge, lane handles 2 floats of the 64-wide row.
// Weight table (wrows x 64 fp32, contiguous) is DMA'd into LDS by the TDM
// (tensor_load_to_lds, waited with s_wait_tensorcnt) once per persistent
// workgroup. Source row selected from src0 (row<split) or src1 (row-split).
// dst[head] += src[tail] * w[type];  cnt[head] += 1 (if cnt != null).
// ---------------------------------------------------------------------------
__global__ __launch_bounds__(256) void edge_scatter_kernel(
    const int* __restrict__ eidx, const int* __restrict__ etype, int type_bias,
    const float* __restrict__ wtab, int wrows,
    const float* __restrict__ src0, const float* __restrict__ src1, int split,
    float* __restrict__ dst, float* __restrict__ cnt, int nedges) {
  __shared__ float lds_w[32 * DD];  // 8KB, covers both 16- and 32-row tables

#if defined(__HIP_DEVICE_COMPILE__) && \
    __has_builtin(__builtin_amdgcn_tensor_load_to_lds) && \
    __has_builtin(__builtin_amdgcn_s_wait_tensorcnt)
  if ((threadIdx.x >> 5) == 0) {
    // Build the Tensor DMA descriptor (D#) for a 2D tile:
    //   data_size=4B, tensor = wrows x 64, tile = wrows x 64, stride = 64.
    unsigned long long ga = (unsigned long long)(uintptr_t)wtab;
    unsigned int lds_off = (unsigned int)(uintptr_t)(&lds_w[0]);  // low32 = LDS offset
    u32x4 g0 = {1u /*count=1, user-mode*/, lds_off,
                (unsigned int)(ga & 0xffffffffull),
                (unsigned int)((ga >> 32) & 0x1ffffffull) | (2u << 30) /*type=2*/};
    i32x8v g1 = {(2 << 16) /*data_size=4B, wg_mask=0*/,
                 (64 << 16) /*tensor_dim0 lo16*/,
                 (wrows << 16) /*tensor_dim1 lo16*/,
                 (64 << 16) /*tile_dim0*/,
                 wrows /*tile_dim1 (tile_dim2=0)*/,
                 64 /*tensor_dim0_stride lo32*/, 0, 0};
    i32x4v g2 = {0, 0, 0, 0};
    i32x4v g3 = {0, 0, 0, 0};
#if __clang_major__ >= 23
    i32x8v g4 = {0, 0, 0, 0, 0, 0, 0, 0};
    __builtin_amdgcn_tensor_load_to_lds(g0, g1, g2, g3, g4, 0);
#else
    __builtin_amdgcn_tensor_load_to_lds(g0, g1, g2, g3, 0);
#endif
    __builtin_amdgcn_s_wait_tensorcnt(0);
  }
  __syncthreads();
#else
  for (int i = threadIdx.x; i < wrows * DD; i += 256) lds_w[i] = wtab[i];
  __syncthreads();
#endif

  const int lane = threadIdx.x & 31;
  const int wave = (blockIdx.x * 256 + threadIdx.x) >> 5;
  const int nwaves = (gridDim.x * 256) >> 5;

  for (int e = wave; e < nedges; e += nwaves) {
    if (lane == 0) {  // stream-ahead prefetch of the edge arrays
      __builtin_prefetch(eidx + e + nwaves, 0, 3);
      __builtin_prefetch(eidx + nedges + e + nwaves, 0, 3);
      __builtin_prefetch(etype + e + nwaves, 0, 3);
    }
    int head = eidx[e];
    int tail = eidx[nedges + e];
    int t = etype[e] - type_bias;
    const float* srow = (tail < split) ? (src0 + (size_t)tail * DD)
                                       : (src1 + (size_t)(tail - split) * DD);
    float2 v = *(const float2*)(srow + 2 * lane);                 // coalesced 256B
    float2 w = *(const float2*)(&lds_w[t * DD + 2 * lane]);       // LDS broadcast
    float* d = dst + (size_t)head * DD + 2 * lane;
    atomic_add_f32(d, v.x * w.x);
    atomic_add_f32(d + 1, v.y * w.y);
    if (cnt != nullptr && lane == 0) atomic_add_f32(cnt + head, 1.0f);
  }
}

// ---------------------------------------------------------------------------
// user_sum[rows[k]] += vals[k] * ent_mean[cols[k]]   (one wave per nnz)
// ---------------------------------------------------------------------------
__global__ __launch_bounds__(256) void interact_scatter_kernel(
    const int* __restrict__ idx, const float* __restrict__ vals,
    const float* __restrict__ ent, float* __restrict__ usum, int nnz) {
  const int lane = threadIdx.x & 31;
  const int wave = (blockIdx.x * 256 + threadIdx.x) >> 5;
  const int nwaves = (gridDim.x * 256) >> 5;
  for (int k = wave; k < nnz; k += nwaves) {
    if (lane == 0) {
      __builtin_prefetch(idx + k + nwaves, 0, 3);
      __builtin_prefetch(idx + nnz + k + nwaves, 0, 3);
    }
    int r = idx[k];
    int c = idx[nnz + k];
    float val = vals[k];
    float2 v = *(const float2*)(ent + (size_t)c * DD + 2 * lane);
    float* d = usum + (size_t)r * DD + 2 * lane;
    atomic_add_f32(d, val * v.x);
    atomic_add_f32(d + 1, val * v.y);
  }
}

// ---------------------------------------------------------------------------
// ent_sum /= max(cnt,1)  (the pre-norm mean is needed by the user scatter)
// ---------------------------------------------------------------------------
__global__ __launch_bounds__(256) void divide_rows_kernel(
    float* __restrict__ sum, const float* __restrict__ cnt, int nrows) {
  int i = blockIdx.x * 256 + threadIdx.x;  // float4 index
  if (i >= nrows * (DD / 4)) return;
  float inv = 1.0f / fmaxf(cnt[i >> 4], 1.0f);
  float4* p = (float4*)sum + i;
  float4 v = *p;
  v.x *= inv; v.y *= inv; v.z *= inv; v.w *= inv;
  *p = v;
}

// ---------------------------------------------------------------------------
// L2-normalize 16 rows per wave via v_wmma_f32_16x16x4_f32.
// Lane L loads float2 x[c] = src[r0+(L&15)][4c + 2*(L>>4) .. +1], c=0..15.
// That register image is simultaneously a valid 16x4 f32 A-tile and 4x16 f32
// B-tile of the SAME data, so wmma(a,a,acc) accumulates the Gram matrix
// X·X^T; its diagonal is the squared row norms. Diagonal of C/D layout:
// (m,m) lives in VGPR m lane m for m<8 and VGPR m-8 lane m+16 for m>=8.
// Then: optionally write normalized rows in place, += into a residual
// buffer, and copy the first copy_rows rows (next hop's "u").
// nrows must cover full waves of 16 rows (all our sizes are multiples of 16).
// ---------------------------------------------------------------------------
__global__ __launch_bounds__(256) void l2norm16_kernel(
    const float* __restrict__ src, float* __restrict__ inplace_out,
    float* __restrict__ acc_out, float* __restrict__ copy_out, int copy_rows,
    int nrows) {
  __shared__ float snorm[8 * 16];
  const int lane = threadIdx.x & 31;
  const int wib = threadIdx.x >> 5;
  const int r0 = (blockIdx.x * 8 + wib) * 16;
  if (r0 >= nrows) return;  // wave-uniform exit: EXEC stays all-ones for WMMA

  const int m = lane & 15;
  const int h = lane >> 4;
  const size_t row = (size_t)(r0 + m);
  const float* p = src + row * DD + 2 * h;
  float2 x[16];
#pragma unroll
  for (int c = 0; c < 16; ++c) x[c] = *(const float2*)(p + 4 * c);

#if defined(__HIP_DEVICE_COMPILE__) && \
    __has_builtin(__builtin_amdgcn_wmma_f32_16x16x4_f32)
  v8f acc = {0.f, 0.f, 0.f, 0.f, 0.f, 0.f, 0.f, 0.f};
#pragma unroll
  for (int c = 0; c < 16; ++c) {
    v2f a;
    a.x = x[c].x;
    a.y = x[c].y;
    acc = __builtin_amdgcn_wmma_f32_16x16x4_f32(false, a, false, a, (short)0,
                                                acc, false, false);
  }
#pragma unroll
  for (int v = 0; v < 8; ++v) {   // scatter the Gram diagonal to LDS
    if (lane == v) snorm[wib * 16 + v] = acc[v];
    if (lane == v + 24) snorm[wib * 16 + 8 + v] = acc[v];
  }
  float ss = snorm[wib * 16 + m];
#else
  float partial = 0.f;
#pragma unroll
  for (int c = 0; c < 16; ++c) partial += x[c].x * x[c].x + x[c].y * x[c].y;
  float ss = partial + __shfl_xor(partial, 16, 32);
#endif

  float scale = 1.0f / fmaxf(sqrtf(ss), 1e-12f);

  if (inplace_out) {
    float* q = inplace_out + row * DD + 2 * h;
#pragma unroll
    for (int c = 0; c < 16; ++c) {
      float2 y = make_float2(x[c].x * scale, x[c].y * scale);
      *(float2*)(q + 4 * c) = y;
    }
  }
  if (acc_out) {
    float* q = acc_out + row * DD + 2 * h;
#pragma unroll
    for (int c = 0; c < 16; ++c) {
      float2 y = *(const float2*)(q + 4 * c);
      y.x += x[c].x * scale;
      y.y += x[c].y * scale;
      *(float2*)(q + 4 * c) = y;
    }
  }
  if (copy_out && (int)row < copy_rows) {
    float* q = copy_out + row * DD + 2 * h;
#pragma unroll
    for (int c = 0; c < 16; ++c) {
      float2 y = make_float2(x[c].x * scale, x[c].y * scale);
      *(float2*)(q + 4 * c) = y;
    }
  }
}

// ---------------------------------------------------------------------------
// Host orchestration
// d_out layout (floats): [0,NU*D) user_res | [NU*D,NN*D) entity_final |
//                        [NN*D,2*NN*D) node_res
// Workspace: entA | node_sum | user_sum | cur_u | ent_cnt  (~128.4 MB)
// ---------------------------------------------------------------------------
extern "C" void kernel_launch(void* const* d_in, const int* in_sizes, int n_in,
                              void* d_out, int out_size, void* d_ws,
                              size_t ws_size, hipStream_t stream) {
  (void)in_sizes; (void)n_in; (void)out_size; (void)ws_size;

  const float* user0 = (const float*)d_in[0];
  const float* ent0 = (const float*)d_in[1];
  const float* weight = (const float*)d_in[2];    // 16 x 64
  const float* xweight = (const float*)d_in[3];   // 32 x 64
  const float* ivals = (const float*)d_in[4];
  const int* eidx = (const int*)d_in[5];          // 2 x EKG
  const int* etype = (const int*)d_in[6];
  const int* xeidx = (const int*)d_in[7];         // 2 x EPREF
  const int* xetype = (const int*)d_in[8];
  const int* iidx = (const int*)d_in[9];          // 2 x NNZ

  float* out_user = (float*)d_out;
  float* out_ent = out_user + (size_t)NU * DD;
  float* out_node = out_user + (size_t)NN * DD;

  float* ws = (float*)d_ws;
  float* entA = ws;                                   // NE*D
  float* node_sum = entA + (size_t)NE * DD;           // NN*D
  float* user_sum = node_sum + (size_t)NN * DD;       // NU*D
  float* cur_u = user_sum + (size_t)NU * DD;          // NU*D
  float* ent_cnt = cur_u + (size_t)NU * DD;           // NE

  // residual bases: node_res = concat(user0, ent0); user_res = user0
  init_out_kernel<<<(NN * (DD / 4) + 255) / 256, 256, 0, stream>>>(
      user0, ent0, out_user, out_node);

  for (int hop = 0; hop < 2; ++hop) {
    // Ping-pong: hop0 accumulates entities into workspace entA; hop1
    // accumulates directly into the entity slice of d_out (in-place
    // mean -> user scatter -> in-place l2norm yields the final output).
    float* ent_sum = (hop == 0) ? entA : out_ent;
    const float* ent_src = (hop == 0) ? ent0 : entA;
    const float* u_src = (hop == 0) ? user0 : cur_u;

    hipMemsetAsync(ent_sum, 0, (size_t)NE * DD * sizeof(float), stream);
    hipMemsetAsync(ent_cnt, 0, (size_t)NE * sizeof(float), stream);
    hipMemsetAsync(node_sum, 0, (size_t)NN * DD * sizeof(float), stream);
    hipMemsetAsync(user_sum, 0, (size_t)NU * DD * sizeof(float), stream);

    // KG scatter: ent_sum[head] += ent_src[tail] * weight[type-1]
    edge_scatter_kernel<<<2048, 256, 0, stream>>>(
        eidx, etype, /*bias=*/1, weight, /*wrows=*/16, ent_src, ent_src,
        /*split=*/NE, ent_sum, ent_cnt, EKG);
    // Pref scatter over concat(u, ent): node_sum[head] += all[tail]*xw[type]
    // (node counts cancel under L2 normalization -> no count buffer needed)
    edge_scatter_kernel<<<2048, 256, 0, stream>>>(
        xeidx, xetype, /*bias=*/0, xweight, /*wrows=*/32, u_src, ent_src,
        /*split=*/NU, node_sum, nullptr, EPREF);
    // entity mean (pre-norm) feeds the user interaction scatter
    divide_rows_kernel<<<(NE * (DD / 4) + 255) / 256, 256, 0, stream>>>(
        ent_sum, ent_cnt, NE);
    interact_scatter_kernel<<<2048, 256, 0, stream>>>(iidx, ivals, ent_sum,
                                                      user_sum, NNZI);
    // L2 norms (WMMA Gram-diagonal): entity in place (next hop's gather
    // source / final output), node += into residual + export first NU rows
    // as next hop's u, user += into residual.
    l2norm16_kernel<<<(NE / 16 + 7) / 8, 256, 0, stream>>>(
        ent_sum, ent_sum, nullptr, nullptr, 0, NE);
    l2norm16_kernel<<<(NN / 16 + 7) / 8, 256, 0, stream>>>(
        node_sum, nullptr, out_node, cur_u, NU, NN);
    l2norm16_kernel<<<(NU / 16 + 7) / 8, 256, 0, stream>>>(
        user_sum, nullptr, out_user, nullptr, 0, NU);
  }
}